// GCN_65317862637616
// MI455X (gfx1250) — compile-verified
//
#include <hip/hip_runtime.h>
#include <hip/hip_bf16.h>

typedef float v2f __attribute__((ext_vector_type(2)));
typedef float v8f __attribute__((ext_vector_type(8)));

// ---------------------------------------------------------------------------
// Degree / normalization helpers
// ---------------------------------------------------------------------------
__global__ void fill_ones_kernel(float* __restrict__ p, int n) {
    int i = blockIdx.x * blockDim.x + threadIdx.x;
    if (i < n) p[i] = 1.0f;  // self-loop contributes 1 to every node's degree
}

__global__ void deg_count_kernel(const int* __restrict__ dst,
                                 float* __restrict__ deg, long long e) {
    long long i = (long long)blockIdx.x * blockDim.x + threadIdx.x;
    if (i < e) atomicAdd(&deg[dst[i]], 1.0f);
}

__global__ void dinv_kernel(const float* __restrict__ deg,
                            float* __restrict__ dinv, int n) {
    int i = blockIdx.x * blockDim.x + threadIdx.x;
    if (i < n) {
        float d = deg[i];
        dinv[i] = (d > 0.0f) ? __frsqrt_rn(d) : 0.0f;
    }
}

// ---------------------------------------------------------------------------
// FP32 WMMA GEMM:  C[M x NDIM] = A[M x KDIM] * W[KDIM x NDIM]
// One wave computes a 16-row tile across all NDIM columns using
// V_WMMA_F32_16X16X4_F32.  M must be a multiple of 16 (100000 = 6250*16).
//
// ISA 16x4 f32 A layout : lanes 0-15 hold {K=k,K=k+1} for M=lane,
//                         lanes 16-31 hold {K=k+2,K=k+3} for M=lane-16.
// B (4x16) mirrored     : lanes 0-15 hold rows k,k+1; lanes 16-31 rows k+2,k+3.
// C/D 8-VGPR layout     : VGPR r -> M = r + 8*(lane>>4), N = lane&15.
//
// W is staged in LDS pair-interleaved: W[k][n] -> sW[(k/2)*STRIDE + 2n + (k&1)]
// so each B fragment is ONE aligned ds_load_b64 (no cross-row repacking).
// STRIDE = 2*NDIM + 32 floats => STRIDE % 64 == 32, so the upper half-wave
// (pair-row + 1) hits the opposite 32 LDS banks: all 64 banks, conflict-free.
// ---------------------------------------------------------------------------
template <int KDIM, int NDIM>
__global__ void __launch_bounds__(128)
gemm_wmma_f32_kernel(const float* __restrict__ A, const float* __restrict__ W,
                     float* __restrict__ C, int M) {
    constexpr int STRIDE = 2 * NDIM + 32;          // floats per k-pair row
    __shared__ float sW[(KDIM / 2) * STRIDE];

    for (int i = threadIdx.x; i < KDIM * NDIM; i += blockDim.x) {
        int k = i / NDIM;
        int n = i - k * NDIM;
        sW[(k >> 1) * STRIDE + 2 * n + (k & 1)] = W[i];
    }
    __syncthreads();

    const int wave = threadIdx.x >> 5;
    const int lane = threadIdx.x & 31;
    const int half = lane >> 4;   // 0: lanes 0-15, 1: lanes 16-31
    const int l15  = lane & 15;
    const int tile = blockIdx.x * 4 + wave;
    if (tile * 16 >= M) return;   // uniform per wave -> EXEC stays all-1s

    // Load all A fragments for this 16-row tile once (8B-aligned float2 loads).
    const int row = tile * 16 + l15;
    v2f af[KDIM / 4];
#pragma unroll
    for (int kk = 0; kk < KDIM / 4; ++kk) {
        const float* ap = A + (size_t)row * KDIM + kk * 4 + 2 * half;
        af[kk] = *reinterpret_cast<const v2f*>(ap);
    }

#pragma unroll
    for (int nt = 0; nt < NDIM / 16; ++nt) {
        v8f acc = {};
#pragma unroll
        for (int kk = 0; kk < KDIM / 4; ++kk) {
            // pair-row index for K = kk*4 + 2*half  is  kk*2 + half
            const float* bp = &sW[(kk * 2 + half) * STRIDE + 2 * (nt * 16 + l15)];
            v2f b = *reinterpret_cast<const v2f*>(bp);
            acc = __builtin_amdgcn_wmma_f32_16x16x4_f32(
                /*neg_a=*/false, af[kk], /*neg_b=*/false, b,
                /*c_mod=*/(short)0, acc, /*reuse_a=*/false, /*reuse_b=*/false);
        }
#pragma unroll
        for (int r = 0; r < 8; ++r) {
            C[(size_t)(tile * 16 + r + 8 * half) * NDIM + nt * 16 + l15] = acc[r];
        }
    }
}

// ---------------------------------------------------------------------------
// Aggregation: out[i] = h[i] * dinv[i]^2  (self-loop term, also initializes
// the destination buffer deterministically), then per-edge scatter-add.
// ---------------------------------------------------------------------------
template <int F>
__global__ void selfloop_init_kernel(const float* __restrict__ h,
                                     const float* __restrict__ dinv,
                                     float* __restrict__ out, int n) {
    long long i = (long long)blockIdx.x * blockDim.x + threadIdx.x;
    if (i < (long long)n * F) {
        int node = (int)(i / F);
        float dv = dinv[node];
        out[i] = h[i] * dv * dv;
    }
}

template <int F>
__global__ void scatter_edges_kernel(const float* __restrict__ h,
                                     const int* __restrict__ src,
                                     const int* __restrict__ dst,
                                     const float* __restrict__ dinv,
                                     float* __restrict__ out, long long e) {
    long long gid  = (long long)blockIdx.x * blockDim.x + threadIdx.x;
    long long edge = gid >> 5;
    int lane       = threadIdx.x & 31;
    if (edge >= e) return;
    int s = src[edge];
    int d = dst[edge];
    float norm = dinv[s] * dinv[d];
    const float* hs = h + (size_t)s * F;
    float* od = out + (size_t)d * F;
#pragma unroll
    for (int f = lane; f < F; f += 32) {
        atomicAdd(&od[f], hs[f] * norm);
    }
}

template <int F, bool RELU>
__global__ void bias_act_kernel(float* __restrict__ h,
                                const float* __restrict__ bias, int n) {
    long long i = (long long)blockIdx.x * blockDim.x + threadIdx.x;
    if (i < (long long)n * F) {
        float v = h[i] + bias[(int)(i % F)];
        h[i] = RELU ? fmaxf(v, 0.0f) : v;
    }
}

// ---------------------------------------------------------------------------
// Launch
// ---------------------------------------------------------------------------
extern "C" void kernel_launch(void* const* d_in, const int* in_sizes, int n_in,
                              void* d_out, int out_size, void* d_ws, size_t ws_size,
                              hipStream_t stream) {
    const float* x  = (const float*)d_in[0];
    const int*   ei = (const int*)d_in[1];   // [2, E] int32
    const float* W1 = (const float*)d_in[2];
    const float* b1 = (const float*)d_in[3];
    const float* W2 = (const float*)d_in[4];
    const float* b2 = (const float*)d_in[5];
    float* out = (float*)d_out;

    const int       N = in_sizes[0] / 32;            // 100000
    const long long E = (long long)in_sizes[1] / 2;  // 1600000
    const int* src = ei;
    const int* dst = ei + E;

    // Workspace layout (floats): deg | dinv | hw1 | h1 | hw2
    float* ws   = (float*)d_ws;
    float* deg  = ws;
    float* dinv = deg + N;
    float* hw1  = dinv + N;                  // N x 64  (x @ W1)
    float* h1   = hw1 + (size_t)N * 64;      // N x 64  (aggregated, relu'd)
    float* hw2  = h1 + (size_t)N * 64;       // N x 128 (h1 @ W2)

    const int BT = 256;

    // 1) degree (with self-loops) and normalization
    fill_ones_kernel<<<(N + BT - 1) / BT, BT, 0, stream>>>(deg, N);
    deg_count_kernel<<<(unsigned)((E + BT - 1) / BT), BT, 0, stream>>>(dst, deg, E);
    dinv_kernel<<<(N + BT - 1) / BT, BT, 0, stream>>>(deg, dinv, N);

    // 2) layer 1: GEMM (WMMA fp32) -> self-loop init -> edge scatter -> bias+relu
    {
        int tiles  = N / 16;                       // 6250
        int blocks = (tiles + 3) / 4;
        gemm_wmma_f32_kernel<32, 64><<<blocks, 128, 0, stream>>>(x, W1, hw1, N);
    }
    {
        long long nf = (long long)N * 64;
        selfloop_init_kernel<64><<<(unsigned)((nf + BT - 1) / BT), BT, 0, stream>>>(hw1, dinv, h1, N);
        long long thr = E * 32;
        scatter_edges_kernel<64><<<(unsigned)((thr + BT - 1) / BT), BT, 0, stream>>>(hw1, src, dst, dinv, h1, E);
        bias_act_kernel<64, true><<<(unsigned)((nf + BT - 1) / BT), BT, 0, stream>>>(h1, b1, N);
    }

    // 3) layer 2: GEMM (WMMA fp32) -> self-loop init (into d_out) -> scatter -> bias
    {
        int tiles  = N / 16;
        int blocks = (tiles + 3) / 4;
        gemm_wmma_f32_kernel<64, 128><<<blocks, 128, 0, stream>>>(h1, W2, hw2, N);
    }
    {
        long long nf = (long long)N * 128;
        selfloop_init_kernel<128><<<(unsigned)((nf + BT - 1) / BT), BT, 0, stream>>>(hw2, dinv, out, N);
        long long thr = E * 32;
        scatter_edges_kernel<128><<<(unsigned)((thr + BT - 1) / BT), BT, 0, stream>>>(hw2, src, dst, dinv, out, E);
        bias_act_kernel<128, false><<<(unsigned)((nf + BT - 1) / BT), BT, 0, stream>>>(out, b2, N);
    }
}